// FeedForwardSE3_53523882442964
// MI455X (gfx1250) — compile-verified
//
#include <hip/hip_runtime.h>
#include <math.h>

typedef __attribute__((ext_vector_type(2))) float v2f;
typedef __attribute__((ext_vector_type(8))) float v8f;

#define TOKENS (4 * 4096)
#define DIM 256
#define HID 1024

#define BM 128
#define BN 128
#define BK 16
#define LSTR 20                       // BK + 4 pad: 80B lines -> 16B aligned, bank-conflict free
#define BUFSZ ((BM + BN) * LSTR)      // floats per LDS buffer

// C[M,N] = A[M,K] * B[K,N], fp32, via v_wmma_f32_16x16x4_f32.
// LOADA_X:      A is x with layout [token][k][mcomp] (row t = token*mcomp + mc).
// STORE_UNPACK: C row t scatters to out[token][col][mc] (layout [b,n,d,m]).
template <int LOADA_X, int STORE_UNPACK>
__global__ __launch_bounds__(256) void se3ff_gemm_wmma(
    const float* __restrict__ A, const float* __restrict__ B,
    float* __restrict__ C, int M, int N, int K, int mcomp) {
  __shared__ __attribute__((aligned(16))) float lds[2 * BUFSZ];

  const int tid = threadIdx.x;
  const int lane = tid & 31;
  const int wid = tid >> 5;
  const int l16 = lane & 15;
  const int hlf = lane >> 4;

  const int m0 = blockIdx.x * BM;
  const int n0 = blockIdx.y * BN;

  // 8 waves in a 4(M) x 2(N) grid; each wave: 32 rows x 64 cols = 2x4 accumulators.
  const int Moff = (wid & 3) * 32;
  const int Noff = (wid >> 2) * 64;

  // ---- per-thread tile-load geometry (hoisted out of the K loop) ----
  int rowa[2], kca[2], krb[2], ncb[2];
  size_t abase[2], bcol[2];
#pragma unroll
  for (int p = 0; p < 2; ++p) {
    const int lin = tid + p * 256;  // 0..511 -> 512 float4 per tile
    rowa[p] = lin >> 2;             // A tile row 0..127
    kca[p] = (lin & 3) * 4;         // A tile k 0,4,8,12
    krb[p] = lin >> 5;              // B tile k-row 0..15
    ncb[p] = (lin & 31) * 4;        // B tile col 0..124
    const int grow = m0 + rowa[p];
    if (LOADA_X) {
      const int token = grow / mcomp;
      const int mc = grow - token * mcomp;
      abase[p] = (size_t)token * K * mcomp + mc;  // x[token][k][mc]
    } else {
      abase[p] = (size_t)grow * K;
    }
    bcol[p] = (size_t)(n0 + ncb[p]);
  }

  const v8f vzero = {};
  v8f acc[2][4];
#pragma unroll
  for (int i = 0; i < 2; ++i)
#pragma unroll
    for (int j = 0; j < 4; ++j) acc[i][j] = vzero;

  float4 ra[2], rb[2];

  auto loadGlobal = [&](int kt) {
    const int ktk = kt * BK;
#pragma unroll
    for (int p = 0; p < 2; ++p) {
      if (LOADA_X) {
        const size_t kb = abase[p] + (size_t)(ktk + kca[p]) * mcomp;
        ra[p].x = A[kb];
        ra[p].y = A[kb + (size_t)mcomp];
        ra[p].z = A[kb + (size_t)(2 * mcomp)];
        ra[p].w = A[kb + (size_t)(3 * mcomp)];
      } else {
        ra[p] = *(const float4*)(A + abase[p] + ktk + kca[p]);
      }
      rb[p] = *(const float4*)(B + (size_t)(ktk + krb[p]) * N + bcol[p]);
    }
  };

  auto storeLds = [&](int buf) {
    const int bo = buf * BUFSZ;
#pragma unroll
    for (int p = 0; p < 2; ++p) {
      *(float4*)&lds[bo + rowa[p] * LSTR + kca[p]] = ra[p];  // A: [row][k]
      const float be[4] = {rb[p].x, rb[p].y, rb[p].z, rb[p].w};
#pragma unroll
      for (int e = 0; e < 4; ++e)  // B transposed: [col][k]
        lds[bo + (BM + ncb[p] + e) * LSTR + krb[p]] = be[e];
    }
  };

  auto compute = [&](int buf) {
    const int bo = buf * BUFSZ;
#pragma unroll
    for (int ks = 0; ks < 4; ++ks) {
      const int k0 = ks * 4 + 2 * hlf;  // lane<16 -> K{0,1}, lane>=16 -> K{2,3}
      v2f af[2], bf[4];
#pragma unroll
      for (int i = 0; i < 2; ++i)
        af[i] = *(const v2f*)&lds[bo + (Moff + i * 16 + l16) * LSTR + k0];
#pragma unroll
      for (int j = 0; j < 4; ++j)
        bf[j] = *(const v2f*)&lds[bo + (BM + Noff + j * 16 + l16) * LSTR + k0];
#pragma unroll
      for (int i = 0; i < 2; ++i)
#pragma unroll
        for (int j = 0; j < 4; ++j)
          acc[i][j] = __builtin_amdgcn_wmma_f32_16x16x4_f32(
              false, af[i], false, bf[j], (short)0, acc[i][j], false, false);
    }
  };

  const int nkt = K / BK;
  loadGlobal(0);
  storeLds(0);
  __syncthreads();
  for (int kt = 0; kt < nkt; ++kt) {
    const int cur = kt & 1;
    const bool more = (kt + 1) < nkt;
    if (more) loadGlobal(kt + 1);   // global->regs overlaps WMMA on current buffer
    compute(cur);
    if (more) storeLds(1 - cur);
    __syncthreads();
  }

  // ---- epilogue: C/D layout = VGPR r: lanes0-15 M=r, lanes16-31 M=r+8 ----
#pragma unroll
  for (int i = 0; i < 2; ++i)
#pragma unroll
    for (int j = 0; j < 4; ++j) {
      const int col = n0 + Noff + j * 16 + l16;
#pragma unroll
      for (int r = 0; r < 8; ++r) {
        const int row = m0 + Moff + i * 16 + r + 8 * hlf;
        const float v = acc[i][j][r];
        if (STORE_UNPACK) {
          const int token = row / mcomp;
          const int mc = row - token * mcomp;
          C[(size_t)token * N * mcomp + (size_t)col * mcomp + mc] = v;
        } else {
          C[(size_t)row * N + col] = v;
        }
      }
    }
}

// In-place norm-over-m + exact GELU gating on H[token*m + mc][j], j in [0,HID).
__global__ __launch_bounds__(256) void se3ff_nonlin(
    float* __restrict__ H, const float* __restrict__ scale,
    const float* __restrict__ bias, int mcomp) {
  const int idx = blockIdx.x * 256 + threadIdx.x;  // token*HID + j
  const int token = idx >> 10;
  const int j = idx & (HID - 1);
  const size_t base = (size_t)token * mcomp * HID + j;
  float v[5];
  float ss = 0.f;
  for (int mc = 0; mc < mcomp; ++mc) {
    v[mc] = H[base + (size_t)mc * HID];
    ss += v[mc] * v[mc];
  }
  const float nrm = fmaxf(sqrtf(ss), 1e-12f);
  const float g = fmaf(nrm, scale[j], bias[j]);
  const float gate = 0.5f * g * (1.0f + erff(g * 0.70710678118654752f));
  const float f = gate / nrm;
  for (int mc = 0; mc < mcomp; ++mc) H[base + (size_t)mc * HID] = v[mc] * f;
}

extern "C" void kernel_launch(void* const* d_in, const int* in_sizes, int n_in,
                              void* d_out, int out_size, void* d_ws,
                              size_t ws_size, hipStream_t stream) {
  (void)in_sizes; (void)n_in; (void)out_size; (void)ws_size;
  // setup_inputs() dict order: per degree {x, w_in, scale, bias, w_out}
  const float* x[3]     = {(const float*)d_in[0], (const float*)d_in[5],  (const float*)d_in[10]};
  const float* w_in[3]  = {(const float*)d_in[1], (const float*)d_in[6],  (const float*)d_in[11]};
  const float* scale[3] = {(const float*)d_in[2], (const float*)d_in[7],  (const float*)d_in[12]};
  const float* bias[3]  = {(const float*)d_in[3], (const float*)d_in[8],  (const float*)d_in[13]};
  const float* w_out[3] = {(const float*)d_in[4], (const float*)d_in[9],  (const float*)d_in[14]};

  float* H = (float*)d_ws;            // max 81920 x 1024 fp32 = 336 MB
  float* out = (float*)d_out;
  size_t outOff = 0;
  const int mlist[3] = {1, 3, 5};

  for (int deg = 0; deg < 3; ++deg) {
    const int m = mlist[deg];
    const int M = TOKENS * m;  // 16384/49152/81920 — all multiples of BM

    dim3 g1(M / BM, HID / BN);
    se3ff_gemm_wmma<1, 0><<<g1, 256, 0, stream>>>(x[deg], w_in[deg], H, M, HID, DIM, m);

    se3ff_nonlin<<<(TOKENS * HID) / 256, 256, 0, stream>>>(H, scale[deg], bias[deg], m);

    dim3 g2(M / BM, DIM / BN);
    se3ff_gemm_wmma<0, 1><<<g2, 256, 0, stream>>>(H, w_out[deg], out + outOff, M, DIM, HID, m);

    outOff += (size_t)TOKENS * DIM * m;
  }
}